// CutOut_23098334118465
// MI455X (gfx1250) — compile-verified
//
#include <hip/hip_runtime.h>

// CutOut: (64, 512, 512, 3) fp32, zero a 50x50 per-batch square, pass labels.
// Memory-roofline kernel: ~402 MB traffic -> ~17us at 23.3 TB/s HBM.
//
// Geometry-matched launch: one row = 512*3 floats = 384 float4s = 12 wave32s.
// One block (384 threads) per row makes b/h/center lookups & the band test
// fully scalar (SALU + s_cbranch); only the rare in-band rows do per-lane
// column math. Data path is b128 non-temporal load/store.

#define CUT_HALF 25

typedef float v4f __attribute__((ext_vector_type(4)));

__global__ __launch_bounds__(384) void cutout_copy_kernel(
    const v4f* __restrict__ img,       // input images as float4
    const int* __restrict__ center_h,  // [64]
    const int* __restrict__ center_w,  // [64]
    v4f* __restrict__ out)             // output images as float4
{
    const int rowIdx = blockIdx.x;          // b*512 + h   (scalar)
    const int b      = rowIdx >> 9;         // scalar
    const int h      = rowIdx & 511;        // scalar
    const int tid    = threadIdx.x;         // float4 index within row [0,384)
    const int i      = rowIdx * 384 + tid;  // global float4 index

    // Streaming read: never re-read -> non-temporal, full b128.
    v4f v = __builtin_nontemporal_load(&img[i]);

    // b is uniform across the workgroup -> scalar load + scalar branch.
    const int ch = center_h[b];
    if (h >= ch - CUT_HALF && h < ch + CUT_HALF) {
        const int cw   = center_w[b];       // scalar load
        const int base = tid * 4;           // float index within row
        #pragma unroll
        for (int e = 0; e < 4; ++e) {
            const int w = (base + e) / 3;   // column of this float (C == 3)
            if (w >= cw - CUT_HALF && w < cw + CUT_HALF) {
                v[e] = 0.0f;                // FILL_VALUE
            }
        }
    }

    // Streaming write: non-temporal store, keep L2 for someone who needs it.
    __builtin_nontemporal_store(v, &out[i]);
}

__global__ void copy_labels_kernel(const int* __restrict__ labels,
                                   int* __restrict__ out_labels)
{
    const int t = threadIdx.x;
    if (t < 64) {
        out_labels[t] = labels[t];  // bit-exact int32 pass-through
    }
}

extern "C" void kernel_launch(void* const* d_in, const int* in_sizes, int n_in,
                              void* d_out, int out_size, void* d_ws, size_t ws_size,
                              hipStream_t stream) {
    (void)in_sizes; (void)n_in; (void)out_size; (void)d_ws; (void)ws_size;

    // setup_inputs() order: images, labels, center_h, center_w
    const v4f* img      = (const v4f*)d_in[0];
    const int* labels   = (const int*)d_in[1];
    const int* center_h = (const int*)d_in[2];
    const int* center_w = (const int*)d_in[3];

    float* out_img = (float*)d_out;
    // labels follow the flattened image output (tuple concatenated flat)
    const long long IMG_FLOATS = 64LL * 512 * 512 * 3;   // 50,331,648
    int* out_labels = (int*)(out_img + IMG_FLOATS);

    const int ROWS  = 64 * 512;   // one block per (b, h) row
    const int BLOCK = 384;        // 384 float4s per row = 12 wave32s

    cutout_copy_kernel<<<ROWS, BLOCK, 0, stream>>>(img, center_h, center_w,
                                                   (v4f*)out_img);
    copy_labels_kernel<<<1, 64, 0, stream>>>(labels, out_labels);
}